// RoIAlign_85744727097691
// MI455X (gfx1250) — compile-verified
//
#include <hip/hip_runtime.h>
#include <hip/hip_bf16.h>
#include <cstdint>

// RoIAlign for MI455X (gfx1250).
// Gather-bound kernel on the CDNA5 async global->LDS path:
//   global_load_async_to_lds_b64  (paired-column bilinear corners)
//   s_wait_asynccnt               (double-buffered per-bin pipeline)
// No WMMA by design: AI ~0.05 FLOP/byte; the bottleneck is L2 gather
// transactions, which the b64 pairing halves (16 -> 8 per bin per lane).

#define OUT_HW   7
#define NBINS    49          // 7*7
#define CHANNELS 256
#define FH       100
#define FW       100
#define SP_SCALE 0.125f

// One bilinear sample -> two row-pair gathers (top/bottom) + 4 weights.
// Mirrors the reference clamping/validity semantics exactly; folds the
// 1/4 sample mean and validity into the weights. For the x-clamped case
// the pair base moves to W-2 and weights remap to (0, w_lo), which is
// value-identical (lx==0 => w_hi==0) and keeps the b64 read in-bounds.
__device__ __forceinline__ void sample_params_pair(float y, float x,
    int& oTop, int& oBot,
    float& w00, float& w01, float& w10, float& w11)
{
    bool valid = (y > -1.0f) && (y < (float)FH) && (x > -1.0f) && (x < (float)FW);
    y = fmaxf(y, 0.0f);
    x = fmaxf(x, 0.0f);
    int yl = (int)floorf(y);
    int xl = (int)floorf(x);
    int yh;
    bool ycond = (yl >= FH - 1);
    bool xcond = (xl >= FW - 1);
    if (ycond) { yl = FH - 1; yh = FH - 1; y = (float)yl; } else { yh = yl + 1; }
    if (xcond) { xl = FW - 1; x = (float)xl; }
    float ly = y - (float)yl;
    float lx = x - (float)xl;            // == 0 when xcond
    float hy = 1.0f - ly, hx = 1.0f - lx;
    float s = valid ? 0.25f : 0.0f;      // 1/4 sample mean * validity
    float w1 = hy * hx * s;              // (yl, xl)
    float w2 = hy * lx * s;              // (yl, xl+1)
    float w3 = ly * hx * s;              // (yh, xl)
    float w4 = ly * lx * s;              // (yh, xl+1)

    int xb;
    if (xcond) {
        xb  = FW - 2;                    // pair (W-2, W-1)
        w00 = 0.0f; w01 = w1;            // only F[W-1] matters; w2==w4==0
        w10 = 0.0f; w11 = w3;
    } else {
        xb  = xl;                        // pair (xl, xl+1)
        w00 = w1;   w01 = w2;
        w10 = w3;   w11 = w4;
    }
    oTop = yl * FW + xb;
    oBot = yh * FW + xb;
}

// Per bin: 8 b64 pair-gathers (4 samples x {top,bottom}) + 16 weights.
__device__ __forceinline__ void bin_params(int bin, float y1, float x1,
    float bin_h, float bin_w, int off[8], float w[16])
{
    int oh = bin / OUT_HW;
    int ow = bin - oh * OUT_HW;
#pragma unroll
    for (int sy = 0; sy < 2; ++sy) {
        float y = y1 + (float)oh * bin_h + ((float)sy + 0.5f) * bin_h * 0.5f;
#pragma unroll
        for (int sx = 0; sx < 2; ++sx) {
            float x = x1 + (float)ow * bin_w + ((float)sx + 0.5f) * bin_w * 0.5f;
            int k = (sy * 2 + sx) * 2;   // pair slot index (2 per sample)
            sample_params_pair(y, x, off[k], off[k + 1],
                               w[2 * k], w[2 * k + 1], w[2 * k + 2], w[2 * k + 3]);
        }
    }
}

// Issue one bin's 8 async b64 gathers into LDS (ASYNCcnt += 8, in-order done).
__device__ __forceinline__ void issue_bin(unsigned long long feat_base,
    uint32_t chan_byte_base, const int off[8], uint32_t lds_lane_byte)
{
#pragma unroll
    for (int i = 0; i < 8; ++i) {
        uint32_t goff = chan_byte_base + ((uint32_t)off[i] << 2);
        uint32_t loff = lds_lane_byte + (uint32_t)i * (CHANNELS * 8);
        asm volatile("global_load_async_to_lds_b64 %0, %1, %2"
                     :: "v"(loff), "v"(goff), "s"(feat_base)
                     : "memory");
    }
}

__global__ __launch_bounds__(CHANNELS) void roi_align_async_kernel(
    const float* __restrict__ features, const float* __restrict__ rois,
    float* __restrict__ out, int n_rois)
{
    __shared__ float2 stage[2][8][CHANNELS];   // 32 KB, conflict-free (8B lane stride)

    int roi = blockIdx.x;
    if (roi >= n_rois) return;
    int c = threadIdx.x;

    const float* r = rois + (size_t)roi * 5;
    int   b  = (int)r[0];
    float x1 = r[1] * SP_SCALE;
    float y1 = r[2] * SP_SCALE;
    float x2 = r[3] * SP_SCALE;
    float y2 = r[4] * SP_SCALE;
    float roi_w = fmaxf(x2 - x1, 1.0f);
    float roi_h = fmaxf(y2 - y1, 1.0f);
    float bin_h = roi_h / (float)OUT_HW;
    float bin_w = roi_w / (float)OUT_HW;

    uint32_t chan_byte_base =
        (uint32_t)(b * CHANNELS + c) * (uint32_t)(FH * FW * 4);
    unsigned long long feat_base = (unsigned long long)(uintptr_t)features;

    // Low 32 bits of a generic LDS pointer are the LDS byte offset
    // (aperture encoding, ISA 10.2) -- matches DS / async-to-LDS addressing.
    uint32_t lds_lane0 = (uint32_t)(uintptr_t)&stage[0][0][c];
    uint32_t lds_lane1 = (uint32_t)(uintptr_t)&stage[1][0][c];

    float* out_c = out + (size_t)roi * (CHANNELS * NBINS) + (size_t)c * NBINS;

    int   off[8];
    float w[16];
    bin_params(0, y1, x1, bin_h, bin_w, off, w);
    issue_bin(feat_base, chan_byte_base, off, lds_lane0);

    for (int bin = 0; bin < NBINS; ++bin) {
        int nb = bin + 1;
        int   onx[8];
        float wn[16];
        if (nb < NBINS) {
            bin_params(nb, y1, x1, bin_h, bin_w, onx, wn);
            issue_bin(feat_base, chan_byte_base, onx,
                      (nb & 1) ? lds_lane1 : lds_lane0);
            // 8 outstanding belong to nb; in-order completion => bin's landed.
            asm volatile("s_wait_asynccnt 8" ::: "memory");
        } else {
            asm volatile("s_wait_asynccnt 0" ::: "memory");
        }

        float acc = 0.0f;
        const float2 (*buf)[CHANNELS] = stage[bin & 1];
#pragma unroll
        for (int i = 0; i < 8; ++i) {
            float2 v = buf[i][c];
            acc = fmaf(w[2 * i],     v.x, acc);
            acc = fmaf(w[2 * i + 1], v.y, acc);
        }
        out_c[bin] = acc;

        if (nb < NBINS) {
#pragma unroll
            for (int i = 0; i < 16; ++i) w[i] = wn[i];
        }
    }
}

extern "C" void kernel_launch(void* const* d_in, const int* in_sizes, int n_in,
                              void* d_out, int out_size, void* d_ws, size_t ws_size,
                              hipStream_t stream)
{
    const float* features = (const float*)d_in[0];
    const float* rois     = (const float*)d_in[1];
    float*       out      = (float*)d_out;
    int n_rois = in_sizes[1] / 5;   // rois is (N, 5)

    dim3 grid((unsigned)n_rois);
    dim3 block(CHANNELS);
    hipLaunchKernelGGL(roi_align_async_kernel, grid, block, 0, stream,
                       features, rois, out, n_rois);
}